// RouteOptimizationGNN_21741124452722
// MI455X (gfx1250) — compile-verified
//
#include <hip/hip_runtime.h>
#include <hip/hip_bf16.h>

#define NN      50000
#define EE      800000
#define ETOT    (EE + NN)      // 850000 edges incl. self loops
#define HH      4
#define CC      64
#define HC      256            // H*C
#define FIN     8
#define GK      256            // K dim for all WMMA GEMMs
#define NEG_SLOPE 0.2f
#define USE_TDM 1

typedef float v2f __attribute__((ext_vector_type(2)));
typedef float v8f __attribute__((ext_vector_type(8)));
typedef unsigned int u32x4 __attribute__((ext_vector_type(4)));
typedef int i32x4 __attribute__((ext_vector_type(4)));
typedef int i32x8 __attribute__((ext_vector_type(8)));

__device__ __forceinline__ float lrelu(float v) {
    return v > 0.0f ? v : v * NEG_SLOPE;
}

// monotonic float <-> uint encoding so we can do float segment-max with atomicMax(u32)
__device__ __forceinline__ unsigned enc_f(float f) {
    unsigned u = __float_as_uint(f);
    return (u & 0x80000000u) ? ~u : (u | 0x80000000u);
}
__device__ __forceinline__ float dec_f(unsigned e) {
    return (e & 0x80000000u) ? __uint_as_float(e & 0x7FFFFFFFu) : __uint_as_float(~e);
}

// ---------------------------------------------------------------------------
// Layer-0 GEMM: h[N,256] = x[N,8] @ W0[8,256]   (K=8, too thin for WMMA)
// ---------------------------------------------------------------------------
__global__ __launch_bounds__(256) void k_gemm_in(const float* __restrict__ x,
                                                 const float* __restrict__ W0,
                                                 float* __restrict__ out) {
    __shared__ float xs[FIN];
    int n = blockIdx.x;
    int c = threadIdx.x;
    if (c < FIN) xs[c] = x[n * FIN + c];
    __syncthreads();
    float s = 0.0f;
#pragma unroll
    for (int k = 0; k < FIN; ++k) s += xs[k] * W0[k * HC + c];
    out[(size_t)n * HC + c] = s;
}

// ---------------------------------------------------------------------------
// FP32 WMMA GEMM: D[M,NC] = A[M,256] @ B[256,NC]  (+ compile-time bias/ReLU)
// One 16x16 tile per wave. The 16x256 (16KB) fp32 A tile is pulled into LDS
// by the Tensor Data Mover (wave 0 issues TENSOR_LOAD_TO_LDS with a 2D D#,
// waits on TENSORcnt, then the workgroup barrier releases the other waves).
// NC is a template constant so B loads / D stores use immediate offsets.
// grid.x = M/16 row tiles; (grid.y * wavesPerBlock) covers NC/16 col tiles.
// ---------------------------------------------------------------------------
template <int NC, bool HAS_BIAS, bool RELU>
__global__ __launch_bounds__(256) void k_gemm_wmma(const float* __restrict__ A,
                                                   const float* __restrict__ B,
                                                   float* __restrict__ D,
                                                   const float* __restrict__ bias) {
    __shared__ float ldsA[16 * GK];           // sole LDS object -> LDS offset 0
    const int row0 = blockIdx.x * 16;
    const int tid  = threadIdx.x;
    const int nw   = blockDim.x >> 5;

#if USE_TDM
    if ((tid >> 5) == 0) {
        // ---- Tensor DMA descriptor (2D tile: 256 x 16 rows of fp32) ----
        const unsigned long long ga =
            (unsigned long long)(const void*)(A + (size_t)row0 * GK);
        u32x4 g0;
        g0[0] = 1u;                                    // count=1, user descriptor
        g0[1] = 0u;                                    // lds_addr = 0 (ldsA)
        g0[2] = (unsigned)(ga & 0xFFFFFFFFu);          // global_addr[31:0]
        g0[3] = (unsigned)((ga >> 32) & 0x01FFFFFFu)   // global_addr[56:32]
              | (2u << 30);                            // type = 2 ("image")
        i32x8 g1;
        g1[0] = 0x00020000;                            // data_size = 4 bytes
        g1[1] = (int)(256u << 16);                     // tensor_dim0 = 256
        g1[2] = (int)(16u << 16);                      // tensor_dim1 = 16
        g1[3] = (int)(256u << 16);                     // tile_dim0 = 256
        g1[4] = 16;                                    // tile_dim1 = 16
        g1[5] = 256;                                   // tensor_dim0_stride = 256
        g1[6] = 0;
        g1[7] = 0;
        i32x4 z4 = {0, 0, 0, 0};
#if defined(__clang_major__) && (__clang_major__ >= 23)
        i32x8 z8 = {0, 0, 0, 0, 0, 0, 0, 0};
        __builtin_amdgcn_tensor_load_to_lds(g0, g1, z4, z4, z8, 0);
#else
        __builtin_amdgcn_tensor_load_to_lds(g0, g1, z4, z4, 0);
#endif
        __builtin_amdgcn_s_wait_tensorcnt(0);
    }
    __syncthreads();
#else
    for (int i = tid; i < 16 * GK; i += blockDim.x) {
        int r = i >> 8;
        int c = i & (GK - 1);
        ldsA[i] = A[(size_t)(row0 + r) * GK + c];
    }
    __syncthreads();
#endif

    const int wave = tid >> 5;
    const int lane = tid & 31;
    const int col0 = (blockIdx.y * nw + wave) * 16;
    if (col0 >= NC) return;                     // wave-uniform

    const int lh  = lane & 15;                  // row-within-tile (A) / col (B,C)
    const int sel = lane >> 4;                  // K-pair select (ISA A 16x4 layout)

    // base pointers: all loop / epilogue offsets become compile-time immediates
    const float* __restrict__ Bp = B + col0 + lh + (size_t)(sel * 2) * NC;
    const float* lp = &ldsA[lh * GK + sel * 2];

    v8f acc = {};
#pragma unroll 4
    for (int k0 = 0; k0 < GK; k0 += 4) {
        // A fragment: lanes 0-15 -> {K=k0,k0+1}, lanes 16-31 -> {K=k0+2,k0+3}
        v2f a = *(const v2f*)(lp + k0);
        // B fragment: same half-wave K pairing, cols striped across lanes
        v2f b;
        b.x = Bp[(size_t)k0 * NC];
        b.y = Bp[(size_t)(k0 + 1) * NC];
        acc = __builtin_amdgcn_wmma_f32_16x16x4_f32(
            /*neg_a=*/false, a, /*neg_b=*/false, b,
            /*c_mod=*/(short)0, acc, /*reuse_a=*/false, /*reuse_b=*/false);
    }

    // C/D layout: VGPR r, lanes 0-15 -> M=r, lanes 16-31 -> M=r+8
    float* __restrict__ Dp = D + (size_t)(row0 + sel * 8) * NC + col0 + lh;
#pragma unroll
    for (int r = 0; r < 8; ++r) {
        float v = acc[r];
        if constexpr (HAS_BIAS) v += bias[col0 + lh];
        if constexpr (RELU) v = fmaxf(v, 0.0f);
        Dp[(size_t)r * NC] = v;
    }
}

// ---------------------------------------------------------------------------
// Per-node attention logits: al_s[n,h] = <h[n,h,:], a_src[h,:]>, same for dst.
// One wave per node; lane covers 8 contiguous features (head = lane/8).
// ---------------------------------------------------------------------------
__global__ __launch_bounds__(256) void k_alphas(const float* __restrict__ h,
                                                const float* __restrict__ a_src,
                                                const float* __restrict__ a_dst,
                                                float* __restrict__ al_s,
                                                float* __restrict__ al_d) {
    int wave = blockIdx.x * (blockDim.x >> 5) + (threadIdx.x >> 5);
    if (wave >= NN) return;
    int lane = threadIdx.x & 31;
    int head = lane >> 3;
    int off  = lane * 8;                      // head*64 + (lane%8)*8
    const float* hr = h + (size_t)wave * HC + off;
    float ps = 0.0f, pd = 0.0f;
#pragma unroll
    for (int i = 0; i < 8; ++i) {
        float hv = hr[i];
        ps += hv * a_src[off + i];
        pd += hv * a_dst[off + i];
    }
#pragma unroll
    for (int m = 4; m >= 1; m >>= 1) {
        ps += __shfl_xor(ps, m, 32);
        pd += __shfl_xor(pd, m, 32);
    }
    if ((lane & 7) == 0) {
        al_s[wave * HH + head] = ps;
        al_d[wave * HH + head] = pd;
    }
}

// ---------------------------------------------------------------------------
// Zero the aggregation target + init segment max (encoded -inf == 0) and den.
// ---------------------------------------------------------------------------
__global__ __launch_bounds__(256) void k_init(float* __restrict__ bufO,
                                              unsigned* __restrict__ m_enc,
                                              float* __restrict__ den) {
    int i = blockIdx.x * blockDim.x + threadIdx.x;
    if (i < NN * HC) bufO[i] = 0.0f;
    if (i < NN * HH) { m_enc[i] = 0u; den[i] = 0.0f; }
}

// ---------------------------------------------------------------------------
// Edge pass 1: segment max of leaky_relu(al_s[src] + al_d[dst]) per head.
// ---------------------------------------------------------------------------
__global__ __launch_bounds__(256) void k_edge_max(const int* __restrict__ ei,
                                                  const float* __restrict__ al_s,
                                                  const float* __restrict__ al_d,
                                                  unsigned* __restrict__ m_enc) {
    int e = blockIdx.x * blockDim.x + threadIdx.x;
    if (e >= ETOT) return;
    int s, d;
    if (e < EE) { s = ei[e]; d = ei[EE + e]; } else { s = d = e - EE; }
    float4 as4 = *(const float4*)(al_s + (size_t)s * HH);
    float4 ad4 = *(const float4*)(al_d + (size_t)d * HH);
    atomicMax(&m_enc[d * HH + 0], enc_f(lrelu(as4.x + ad4.x)));
    atomicMax(&m_enc[d * HH + 1], enc_f(lrelu(as4.y + ad4.y)));
    atomicMax(&m_enc[d * HH + 2], enc_f(lrelu(as4.z + ad4.z)));
    atomicMax(&m_enc[d * HH + 3], enc_f(lrelu(as4.w + ad4.w)));
}

// ---------------------------------------------------------------------------
// Edge pass 2: ee = exp(e - m[dst]); den[dst] += ee  (segment softmax denom)
// ---------------------------------------------------------------------------
__global__ __launch_bounds__(256) void k_edge_exp(const int* __restrict__ ei,
                                                  const float* __restrict__ al_s,
                                                  const float* __restrict__ al_d,
                                                  const unsigned* __restrict__ m_enc,
                                                  float* __restrict__ den,
                                                  float* __restrict__ ee) {
    int e = blockIdx.x * blockDim.x + threadIdx.x;
    if (e >= ETOT) return;
    int s, d;
    if (e < EE) { s = ei[e]; d = ei[EE + e]; } else { s = d = e - EE; }
    float4 as4 = *(const float4*)(al_s + (size_t)s * HH);
    float4 ad4 = *(const float4*)(al_d + (size_t)d * HH);
    float4 ev;
    ev.x = __expf(lrelu(as4.x + ad4.x) - dec_f(m_enc[d * HH + 0]));
    ev.y = __expf(lrelu(as4.y + ad4.y) - dec_f(m_enc[d * HH + 1]));
    ev.z = __expf(lrelu(as4.z + ad4.z) - dec_f(m_enc[d * HH + 2]));
    ev.w = __expf(lrelu(as4.w + ad4.w) - dec_f(m_enc[d * HH + 3]));
    atomicAdd(&den[d * HH + 0], ev.x);
    atomicAdd(&den[d * HH + 1], ev.y);
    atomicAdd(&den[d * HH + 2], ev.z);
    atomicAdd(&den[d * HH + 3], ev.w);
    *(float4*)(ee + (size_t)e * HH) = ev;
}

// ---------------------------------------------------------------------------
// Edge pass 3: out[dst] += h[src] * alpha.  One wave per edge; each lane
// handles 8 of the 256 features (head = lane/8), alpha = ee/den per head.
// ---------------------------------------------------------------------------
__global__ __launch_bounds__(256) void k_edge_aggr(const int* __restrict__ ei,
                                                   const float* __restrict__ h,
                                                   const float* __restrict__ ee,
                                                   const float* __restrict__ den,
                                                   float* __restrict__ out) {
    int wave = blockIdx.x * (blockDim.x >> 5) + (threadIdx.x >> 5);
    if (wave >= ETOT) return;
    int lane = threadIdx.x & 31;
    int s, d;
    if (wave < EE) { s = ei[wave]; d = ei[EE + wave]; } else { s = d = wave - EE; }
    int head = lane >> 3;
    float alpha = ee[(size_t)wave * HH + head] / den[d * HH + head];
    const float* hs = h + (size_t)s * HC + lane * 8;
    float* od = out + (size_t)d * HC + lane * 8;
    float4 v0 = *(const float4*)(hs);
    float4 v1 = *(const float4*)(hs + 4);
    atomicAdd(&od[0], v0.x * alpha);
    atomicAdd(&od[1], v0.y * alpha);
    atomicAdd(&od[2], v0.z * alpha);
    atomicAdd(&od[3], v0.w * alpha);
    atomicAdd(&od[4], v1.x * alpha);
    atomicAdd(&od[5], v1.y * alpha);
    atomicAdd(&od[6], v1.z * alpha);
    atomicAdd(&od[7], v1.w * alpha);
}

// ---------------------------------------------------------------------------
// Bias add (+ optional ReLU) epilogue over [N,256]
// ---------------------------------------------------------------------------
__global__ __launch_bounds__(256) void k_bias_act(float* __restrict__ buf,
                                                  const float* __restrict__ bias,
                                                  int do_relu) {
    int i = blockIdx.x * blockDim.x + threadIdx.x;
    if (i >= NN * HC) return;
    float v = buf[i] + bias[i & (HC - 1)];
    if (do_relu) v = fmaxf(v, 0.0f);
    buf[i] = v;
}

// ---------------------------------------------------------------------------
// Final heads: out[n] = <hid_o[n,:64], ow2> + ob2 ; out[N+n] = time analog.
// One wave per node, 2 features per lane, full-wave shuffle reduction.
// ---------------------------------------------------------------------------
__global__ __launch_bounds__(256) void k_final(const float* __restrict__ hid_o,
                                               const float* __restrict__ hid_t,
                                               const float* __restrict__ ow2,
                                               const float* __restrict__ ob2,
                                               const float* __restrict__ tw2,
                                               const float* __restrict__ tb2,
                                               float* __restrict__ out) {
    int n = blockIdx.x * (blockDim.x >> 5) + (threadIdx.x >> 5);
    if (n >= NN) return;
    int lane = threadIdx.x & 31;
    float2 ho = *(const float2*)(hid_o + (size_t)n * CC + lane * 2);
    float2 ht = *(const float2*)(hid_t + (size_t)n * CC + lane * 2);
    float so = ho.x * ow2[2 * lane] + ho.y * ow2[2 * lane + 1];
    float st = ht.x * tw2[2 * lane] + ht.y * tw2[2 * lane + 1];
#pragma unroll
    for (int m = 16; m >= 1; m >>= 1) {
        so += __shfl_xor(so, m, 32);
        st += __shfl_xor(st, m, 32);
    }
    if (lane == 0) {
        out[n]      = so + ob2[0];
        out[NN + n] = st + tb2[0];
    }
}

// ---------------------------------------------------------------------------
extern "C" void kernel_launch(void* const* d_in, const int* in_sizes, int n_in,
                              void* d_out, int out_size, void* d_ws, size_t ws_size,
                              hipStream_t stream) {
    (void)in_sizes; (void)n_in; (void)out_size; (void)ws_size;
    const float* x       = (const float*)d_in[0];
    const int*   ei      = (const int*)  d_in[1];
    const float* W0      = (const float*)d_in[2];
    const float* Ws      = (const float*)d_in[3];
    const float* att_src = (const float*)d_in[4];
    const float* att_dst = (const float*)d_in[5];
    const float* bconv   = (const float*)d_in[6];
    const float* ow1     = (const float*)d_in[7];
    const float* ob1     = (const float*)d_in[8];
    const float* ow2     = (const float*)d_in[9];
    const float* ob2     = (const float*)d_in[10];
    const float* tw1     = (const float*)d_in[11];
    const float* tb1     = (const float*)d_in[12];
    const float* tw2     = (const float*)d_in[13];
    const float* tb2     = (const float*)d_in[14];
    float* out = (float*)d_out;

    // workspace carve (~119 MB)
    float*    bufF  = (float*)d_ws;                 // [N,256] GEMM features
    float*    bufO  = bufF + (size_t)NN * HC;       // [N,256] aggregated output
    float*    al_s  = bufO + (size_t)NN * HC;       // [N,4]
    float*    al_d  = al_s + (size_t)NN * HH;       // [N,4]
    unsigned* m_enc = (unsigned*)(al_d + (size_t)NN * HH);  // [N,4]
    float*    den   = (float*)(m_enc + (size_t)NN * HH);    // [N,4]
    float*    ee    = den + (size_t)NN * HH;        // [Etot,4]
    float*    hid_o = bufF;                          // reuse after last GAT layer
    float*    hid_t = bufF + (size_t)NN * CC;

    const int blkElems  = (NN * HC + 255) / 256;     // 50000
    const int blkEdges  = (ETOT + 255) / 256;        // 3321
    const int blkEdgeWv = (ETOT + 7) / 8;            // 106250 (8 waves/block)
    const int blkNodeWv = (NN + 7) / 8;              // 6250

    for (int layer = 0; layer < 3; ++layer) {
        // features = h_in @ W  -> bufF
        if (layer == 0) {
            k_gemm_in<<<NN, 256, 0, stream>>>(x, W0, bufF);
        } else {
            const float* W = Ws + (size_t)(layer - 1) * HC * HC;
            k_gemm_wmma<HC, false, false><<<dim3(NN / 16, 2), 256, 0, stream>>>(
                bufO, W, bufF, nullptr);
        }
        k_alphas<<<blkNodeWv, 256, 0, stream>>>(
            bufF, att_src + (size_t)layer * HC, att_dst + (size_t)layer * HC, al_s, al_d);
        k_init<<<blkElems, 256, 0, stream>>>(bufO, m_enc, den);
        k_edge_max<<<blkEdges, 256, 0, stream>>>(ei, al_s, al_d, m_enc);
        k_edge_exp<<<blkEdges, 256, 0, stream>>>(ei, al_s, al_d, m_enc, den, ee);
        k_edge_aggr<<<blkEdgeWv, 256, 0, stream>>>(ei, bufF, ee, den, bufO);
        k_bias_act<<<blkElems, 256, 0, stream>>>(
            bufO, bconv + (size_t)layer * HC, layer < 2 ? 1 : 0);
    }

    // MLP heads: hid = relu(h @ w1 + b1)   (WMMA, Nc=64, 4 waves/block)
    k_gemm_wmma<CC, true, true><<<dim3(NN / 16, 1), 128, 0, stream>>>(bufO, ow1, hid_o, ob1);
    k_gemm_wmma<CC, true, true><<<dim3(NN / 16, 1), 128, 0, stream>>>(bufO, tw1, hid_t, tb1);
    k_final<<<blkNodeWv, 256, 0, stream>>>(hid_o, hid_t, ow2, ob2, tw2, tb2, out);
}